// INT4SymmetricWeightsDecompressor_7928509629314
// MI455X (gfx1250) — compile-verified
//
#include <hip/hip_runtime.h>
#include <stdint.h>

// INT4 symmetric weight decompressor for MI455X (gfx1250).
// Memory-bound: ~272 MB of one-touch traffic -> ~12us floor @ 23.3 TB/s.
// One wave32 per 4 consecutive 128-value groups. Packed bytes are staged
// through LDS with 4 back-to-back global_load_async_to_lds_b64 (ASYNCcnt),
// sharing one base address via the instruction's immediate offset (which the
// async path adds to BOTH the global and LDS addresses), then a single
// s_wait_asynccnt. Outputs leave via coalesced non-temporal b128 stores.

typedef float v4f __attribute__((ext_vector_type(4)));
typedef int   v2i __attribute__((ext_vector_type(2)));

#define AS1 __attribute__((address_space(1)))
#define AS3 __attribute__((address_space(3)))

#define OUT_ROWS         4096
#define GROUPS_STORED    344    // leading group dim of packed/scale inputs
#define GROUPS_USED      86     // 11008 / 128 groups actually materialized
#define ROW_K            11008
#define GROUPS_PER_WAVE  4
#define WAVES_PER_BLOCK  8
#define GROUPS_PER_BLOCK (GROUPS_PER_WAVE * WAVES_PER_BLOCK)   // 32
#define THREADS          (WAVES_PER_BLOCK * 32)                // 256

#if defined(__AMDGCN__) && __has_builtin(__builtin_amdgcn_global_load_async_to_lds_b64)
#define USE_ASYNC_LDS 1
#endif

__global__ __launch_bounds__(THREADS)
void int4_decompress_kernel(const int* __restrict__ packed,
                            const float* __restrict__ scale,
                            float* __restrict__ out)
{
    const int lane = threadIdx.x & 31;   // wave32
    const int wv   = threadIdx.x >> 5;
    const int g0   = blockIdx.x * GROUPS_PER_BLOCK + wv * GROUPS_PER_WAVE;
    const int row  = blockIdx.y;

#ifdef USE_ASYNC_LDS
    // Per-wave region: GROUPS_PER_WAVE iterations x 256 B, matching the
    // global 256 B/group stride so one immediate offset serves both sides.
    __shared__ v2i smem[THREADS * GROUPS_PER_WAVE];   // 8 KB
#endif

    if (g0 >= GROUPS_USED) return;       // uniform per wave

    int ng = GROUPS_USED - g0;
    if (ng > GROUPS_PER_WAVE) ng = GROUPS_PER_WAVE;  // wave-uniform

    const long rowbase = (long)row * GROUPS_STORED;
    const int* src0    = packed + (rowbase + g0) * 64 + lane * 2;

    // Prefetch the next block's span (global_prefetch_b8). Always in-bounds:
    // g0 + 32 <= 116 < GROUPS_STORED.
    __builtin_prefetch(src0 + GROUPS_PER_BLOCK * 64, 0, 0);

    v2i p[GROUPS_PER_WAVE];
#ifdef USE_ASYNC_LDS
    {
        AS1 v2i* gsrc = (AS1 v2i*)(uintptr_t)src0;
        AS3 v2i* ldst = (AS3 v2i*)(uint32_t)(uintptr_t)(&smem[wv * 128 + lane]);
        // Immediate offset must be a literal constant: unroll by hand.
        __builtin_amdgcn_global_load_async_to_lds_b64(gsrc, ldst, 0, 0);
        if (ng > 1) __builtin_amdgcn_global_load_async_to_lds_b64(gsrc, ldst, 256, 0);
        if (ng > 2) __builtin_amdgcn_global_load_async_to_lds_b64(gsrc, ldst, 512, 0);
        if (ng > 3) __builtin_amdgcn_global_load_async_to_lds_b64(gsrc, ldst, 768, 0);
#if __has_builtin(__builtin_amdgcn_s_wait_asynccnt)
        __builtin_amdgcn_s_wait_asynccnt(0);
#else
        asm volatile("s_wait_asynccnt 0" ::: "memory");
#endif
        asm volatile("" ::: "memory");   // keep the LDS reads below the wait
#pragma unroll
        for (int i = 0; i < GROUPS_PER_WAVE; ++i)
            if (i < ng)
                p[i] = smem[wv * 128 + lane + i * 32];  // +256 B per iteration
    }
#else
    {
#pragma unroll
        for (int i = 0; i < GROUPS_PER_WAVE; ++i)
            if (i < ng)
                p[i] = __builtin_nontemporal_load((const v2i*)(src0 + i * 64));
    }
#endif

    // 4 consecutive scales, 16 B aligned (g0 % 4 == 0, rowbase % 8 == 0);
    // always in-bounds of the 344 stored groups even when ng < 4.
    const v4f sv = *(const v4f*)(scale + rowbase + g0);

    float* dst0 = out + (long)row * ROW_K + g0 * 128 + lane * 4;
#pragma unroll
    for (int i = 0; i < GROUPS_PER_WAVE; ++i) {
        if (i < ng) {
            const float s   = sv[i];
            const float m8s = -8.0f * s;   // (w - 8) * s == fma(w, s, -8s)
            const int b0 = p[i].x;
            const int b1 = p[i].y;
            v4f o;
            o.x = __builtin_fmaf((float)(b0 & 15),        s, m8s);
            o.y = __builtin_fmaf((float)((b0 >> 4) & 15), s, m8s);
            o.z = __builtin_fmaf((float)(b1 & 15),        s, m8s);
            o.w = __builtin_fmaf((float)((b1 >> 4) & 15), s, m8s);
            __builtin_nontemporal_store(o, (v4f*)(dst0 + i * 128));
        }
    }
}

extern "C" void kernel_launch(void* const* d_in, const int* in_sizes, int n_in,
                              void* d_out, int out_size, void* d_ws, size_t ws_size,
                              hipStream_t stream)
{
    const int*   packed = (const int*)d_in[0];
    const float* scale  = (const float*)d_in[1];
    float*       out    = (float*)d_out;

    dim3 grid((GROUPS_USED + GROUPS_PER_BLOCK - 1) / GROUPS_PER_BLOCK, OUT_ROWS);
    int4_decompress_kernel<<<grid, THREADS, 0, stream>>>(packed, scale, out);
}